// YOLOXDarkSSL_48284022341748
// MI455X (gfx1250) — compile-verified
//
#include <hip/hip_runtime.h>
#include <math.h>

typedef __attribute__((ext_vector_type(2))) float v2f;
typedef __attribute__((ext_vector_type(4))) float v4f;
typedef __attribute__((ext_vector_type(8))) float v8f;

#define B_      16
#define NV_     128
#define FD_     8192
#define G_      128
#define TINV_   5.0f          /* 1 / 0.2 */
#define EPS_    1e-6f
#define MARGIN_ 1.0f

/* workspace layout (float offsets) */
#define OFF_INVQ 0
#define OFF_INVK 2048
#define OFF_INVN 4096
#define OFF_SQ   6144
#define OFF_SK   8192
#define OFF_PART 10240
#define OFF_S    16384
#define OFF_L    (16384 + 262144)

/* ---------------- kernel 1: row norms ---------------- */
__global__ __launch_bounds__(256) void rownorm_kernel(const float* __restrict__ q,
                                                      const float* __restrict__ k,
                                                      const float* __restrict__ n,
                                                      float* __restrict__ ws) {
    int bid = blockIdx.x;          // 0..6143
    int tensor = bid >> 11;        // 0:q 1:k 2:n
    int row = bid & 2047;          // b*128 + i
    const float* src = (tensor == 0 ? q : tensor == 1 ? k : n) + (size_t)row * FD_;
    int tid = threadIdx.x;
    float ssq = 0.f, s = 0.f;
    const v4f* src4 = (const v4f*)src;
    for (int p = tid; p < FD_ / 4; p += 256) {
        v4f v = src4[p];
        ssq += v.x * v.x + v.y * v.y + v.z * v.z + v.w * v.w;
        s += v.x + v.y + v.z + v.w;
    }
    __shared__ float r0[256], r1[256];
    r0[tid] = ssq; r1[tid] = s;
    __syncthreads();
    for (int off = 128; off; off >>= 1) {
        if (tid < off) { r0[tid] += r0[tid + off]; r1[tid] += r1[tid + off]; }
        __syncthreads();
    }
    if (tid == 0) {
        float inv = 1.0f / fmaxf(sqrtf(r0[0]), 1e-12f);
        ws[(tensor == 0 ? OFF_INVQ : tensor == 1 ? OFF_INVK : OFF_INVN) + row] = inv;
        if (tensor == 0) ws[OFF_SQ + row] = r1[0] * inv;
        else if (tensor == 1) ws[OFF_SK + row] = r1[0] * inv;
    }
}

/* ---------------- kernel 2: WMMA f32 GEMMs ----------------
 * S[b] = (q[b] @ k[b]^T) scaled, L[b] = (q[b] @ n[b]^T) scaled.
 * grid (8 i-tiles, 2 matrices, 16 batches), 8 waves/block, wave w -> j-tile w. */
#define KC 64
#define ASTR 68   /* row stride in floats: 16B-aligned float4, bank-conflict free */

__global__ __launch_bounds__(256) void gemm_kernel(const float* __restrict__ q,
                                                   const float* __restrict__ k,
                                                   const float* __restrict__ n,
                                                   float* __restrict__ ws) {
    int i16 = blockIdx.x;          // 0..7
    int mat = blockIdx.y;          // 0 -> S(k), 1 -> L(n)
    int b   = blockIdx.z;          // 0..15
    int tid = threadIdx.x;
    int wave = tid >> 5;           // 0..7 == j16
    int lane = tid & 31;
    int half = lane >> 4;          // 0: K=0,1  1: K=2,3
    int lm = lane & 15;
    int koff = half * 2;

    const float* Aptr = q + (size_t)b * NV_ * FD_ + (size_t)(i16 * 16) * FD_;
    const float* Bptr = (mat == 0 ? k : n) + (size_t)b * NV_ * FD_ + (size_t)(wave * 16) * FD_;

    __shared__ float Abuf[16 * ASTR];
    __shared__ float Bbuf[8 * 16 * ASTR];
    float* myB = Bbuf + wave * 16 * ASTR;

    v8f c = {};

    int ar = tid >> 4;             // A row 0..15
    int ac4 = tid & 15;            // float4 column

    for (int ch = 0; ch < FD_; ch += KC) {
        /* A tile 16x64: one float4 per thread, coalesced */
        {
            v4f v = *(const v4f*)(Aptr + (size_t)ar * FD_ + ch + ac4 * 4);
            *(v4f*)(Abuf + ar * ASTR + ac4 * 4) = v;
        }
        /* B tile 16x64 per wave: 8 float4 per lane, coalesced */
        for (int s = 0; s < 8; ++s) {
            int p = s * 32 + lane;
            int br = p >> 4, bc4 = p & 15;
            v4f v = *(const v4f*)(Bptr + (size_t)br * FD_ + ch + bc4 * 4);
            *(v4f*)(myB + br * ASTR + bc4 * 4) = v;
        }
        __syncthreads();
        /* 16 WMMAs per chunk; A frag: M=lane%16, K=koff..koff+1; B frag: N=lane%16 */
        for (int kk = 0; kk < KC; kk += 4) {
            v2f a  = *(const v2f*)(Abuf + lm * ASTR + kk + koff);
            v2f bb = *(const v2f*)(myB  + lm * ASTR + kk + koff);
            c = __builtin_amdgcn_wmma_f32_16x16x4_f32(false, a, false, bb,
                                                      (short)0, c, false, false);
        }
        __syncthreads();
    }

    /* epilogue: apply L2-normalization scales; C/D layout: N=lane%16, M=v+8*half */
    const float* invq = ws + OFF_INVQ + b * NV_;
    const float* invx = ws + (mat == 0 ? OFF_INVK : OFF_INVN) + b * NV_;
    float* Out = ws + (mat == 0 ? OFF_S : OFF_L) + (size_t)b * G_ * G_;
    int jg = wave * 16 + lm;
    float sx = invx[jg];
    for (int v = 0; v < 8; ++v) {
        int ig = i16 * 16 + v + half * 8;
        Out[ig * G_ + jg] = c[v] * invq[ig] * sx;
    }
}

/* ---------------- kernel 3: per-batch loss parts ---------------- */
__global__ __launch_bounds__(256) void loss_kernel(float* __restrict__ ws) {
    int b = blockIdx.x;
    int tid = threadIdx.x;
    const float* S  = ws + OFF_S + (size_t)b * G_ * G_;
    const float* L  = ws + OFF_L + (size_t)b * G_ * G_;
    const float* sq = ws + OFF_SQ + b * NV_;
    const float* sk = ws + OFF_SK + b * NV_;
    const float cc = (float)FD_ * EPS_ * EPS_;

    __shared__ float dqk_d[G_], dkq_d[G_];
    if (tid < G_) {
        float Sii = S[tid * G_ + tid];
        float dq = sq[tid] - sk[tid];
        dqk_d[tid] = sqrtf(fmaxf(2.f - 2.f * Sii + 2.f * EPS_ * dq + cc, 0.f));
        dkq_d[tid] = sqrtf(fmaxf(2.f - 2.f * Sii - 2.f * EPS_ * dq + cc, 0.f));
    }
    __syncthreads();

    float sm = 0.f, tri = 0.f, cyc = 0.f;
    for (int p = tid; p < G_ * G_; p += 256) {
        int i = p >> 7, j = p & 127;
        float Sij = S[i * G_ + j];
        float Sji = S[j * G_ + i];
        float d = Sij - (i == j ? 1.f : 0.f);
        sm += d * d;
        if (i != j) {
            float dqk = sqrtf(fmaxf(2.f - 2.f * Sij + 2.f * EPS_ * (sq[i] - sk[j]) + cc, 0.f));
            float dkq = sqrtf(fmaxf(2.f - 2.f * Sji + 2.f * EPS_ * (sk[i] - sq[j]) + cc, 0.f));
            tri += fmaxf(dqk_d[i] - dqk + MARGIN_, 0.f);
            tri += fmaxf(dkq_d[i] - dkq + MARGIN_, 0.f);
            cyc += fabsf(Sij - Sji);
        }
    }

    /* InfoNCE: one wave32 per row group, 16 rows per wave, 4 negatives per lane */
    int wave = tid >> 5, lane = tid & 31;
    float ce = 0.f;
    for (int r = 0; r < 16; ++r) {
        int i = wave * 16 + r;
        float lpos = S[i * G_ + i] * TINV_;
        float l0 = L[i * G_ + lane]       * TINV_;
        float l1 = L[i * G_ + lane + 32]  * TINV_;
        float l2 = L[i * G_ + lane + 64]  * TINV_;
        float l3 = L[i * G_ + lane + 96]  * TINV_;
        float m = fmaxf(fmaxf(fmaxf(l0, l1), fmaxf(l2, l3)), lpos);
        for (int off = 16; off; off >>= 1) m = fmaxf(m, __shfl_xor(m, off, 32));
        float s = __expf(l0 - m) + __expf(l1 - m) + __expf(l2 - m) + __expf(l3 - m);
        for (int off = 16; off; off >>= 1) s += __shfl_xor(s, off, 32);
        s += __expf(lpos - m);
        if (lane == 0) ce += m + __logf(s) - lpos;
    }

    __shared__ float red[256];
    float vals[4] = { sm, tri, cyc, ce };
    float outv[4];
    for (int qi = 0; qi < 4; ++qi) {
        __syncthreads();
        red[tid] = vals[qi];
        __syncthreads();
        for (int off = 128; off; off >>= 1) {
            if (tid < off) red[tid] += red[tid + off];
            __syncthreads();
        }
        outv[qi] = red[0];
    }
    if (tid == 0) {
        float* part = ws + OFF_PART + b * 4;
        part[0] = outv[0] / (float)(G_ * G_);
        part[1] = outv[1];
        part[2] = outv[2] / (float)(G_ * (G_ - 1));
        part[3] = outv[3] / (float)(G_ * G_);   /* mean over rows, then /g */
    }
}

/* ---------------- kernel 4: final combine (cumsum -> weights) ---------------- */
__global__ void final_kernel(const float* __restrict__ ws, float* __restrict__ out) {
    int lane = threadIdx.x;
    float v = 0.f;
    if (lane < B_) {
        const float* p = ws + OFF_PART + lane * 4;
        float tri_cnt = 2.0f * G_ * (G_ - 1);
        /* sum_b cumsum(tri)[b] == sum_b tri_b * (B - b) */
        v = p[0] + p[2] + p[3] + p[1] * (float)(B_ - lane) / tri_cnt;
    }
    for (int off = 16; off; off >>= 1) v += __shfl_xor(v, off, 32);
    if (lane == 0) out[0] = v;
}

extern "C" void kernel_launch(void* const* d_in, const int* in_sizes, int n_in,
                              void* d_out, int out_size, void* d_ws, size_t ws_size,
                              hipStream_t stream) {
    (void)in_sizes; (void)n_in; (void)out_size; (void)ws_size;
    const float* q = (const float*)d_in[0];
    const float* k = (const float*)d_in[1];
    const float* n = (const float*)d_in[2];
    float* ws = (float*)d_ws;
    float* out = (float*)d_out;

    rownorm_kernel<<<3 * B_ * NV_, 256, 0, stream>>>(q, k, n, ws);
    gemm_kernel<<<dim3(8, 2, B_), 256, 0, stream>>>(q, k, n, ws);
    loss_kernel<<<B_, 256, 0, stream>>>(ws);
    final_kernel<<<1, 32, 0, stream>>>(ws, out);
}